// DDK_word_dim_77644418777651
// MI455X (gfx1250) — compile-verified
//
#include <hip/hip_runtime.h>
#include <float.h>
#include <stdint.h>

// Problem shape (fixed by the reference setup): input (l=32, n=2048, m=1024) fp32.
#define L_DIM 32
#define N_DIM 2048
#define M_DIM 1024

// ---------------- block-wide min/max reduction (wave32) ----------------
__device__ inline void block_minmax_reduce(float& mn, float& mx) {
#pragma unroll
  for (int off = 16; off > 0; off >>= 1) {
    mn = fminf(mn, __shfl_xor(mn, off, 32));
    mx = fmaxf(mx, __shfl_xor(mx, off, 32));
  }
  __shared__ float smn[32];
  __shared__ float smx[32];
  const int wid  = threadIdx.x >> 5;
  const int lane = threadIdx.x & 31;
  if (lane == 0) { smn[wid] = mn; smx[wid] = mx; }
  __syncthreads();
  if (threadIdx.x == 0) {
    const int nw = (blockDim.x + 31) >> 5;
    for (int i = 1; i < nw; ++i) { mn = fminf(mn, smn[i]); mx = fmaxf(mx, smx[i]); }
  }
}

// ---------------- stage 1: streaming min/max via async global->LDS ----------------
// 1024 blocks x 256 threads x 64 iters x float4 = 64M floats.
// 8-slot LDS ring, issued in groups of 4 => up to 8 async b128 loads in flight
// per wave (~32MB in flight device-wide; enough to cover HBM latency at 23.3TB/s).
// Walks the tensor BACK-to-FRONT in 4KB stripes so the first ~192MB of the
// tensor is what survives in the 192MB L2 for the recurrence kernel.
#define R_BLOCKS 1024
#define R_THREADS 256
#define R_ITER 64
#define R_CHUNKS 16  // 16 chunks x 4 iters

#define ASYNC_B128(ldsoff, gaddr)                                   \
  asm volatile("global_load_async_to_lds_b128 %0, %1, off"          \
               :: "v"(ldsoff), "v"(gaddr) : "memory")

__global__ void __launch_bounds__(R_THREADS)
minmax_stage1(const float* __restrict__ in, float* __restrict__ ws) {
  const int tid = threadIdx.x;
  const int b   = blockIdx.x;

  __shared__ float4 buf[8][R_THREADS];           // 32KB ring
  const uint32_t l0 = (uint32_t)(uintptr_t)(&buf[0][tid]);
  const float4* gin = (const float4*)in;

  // global iter g in [0,64): data chunk = (63-g)*R_BLOCKS + b (descending stripes)
  auto gaddr = [&](int g) {
    return (uint64_t)(uintptr_t)(
        gin + ((uint64_t)(R_ITER - 1 - g) * R_BLOCKS + (uint64_t)b) * R_THREADS + tid);
  };

  // prologue: chunk 0 -> slots 0..3
#pragma unroll
  for (int j = 0; j < 4; ++j) ASYNC_B128(l0 + (uint32_t)(j * 4096), gaddr(j));

  float mn = FLT_MAX, mx = -FLT_MAX;
#pragma unroll
  for (int c = 0; c < R_CHUNKS; ++c) {
    const int cur = (c & 1) ? 4 : 0;
    if (c + 1 < R_CHUNKS) {
      const int nxt = cur ^ 4;
      // WAR guard: our LDS reads of the slots being re-targeted (2 chunks old) drained
      asm volatile("s_wait_dscnt 0x0" ::: "memory");
#pragma unroll
      for (int j = 0; j < 4; ++j)
        ASYNC_B128(l0 + (uint32_t)((nxt + j) * 4096), gaddr((c + 1) * 4 + j));
      // async loads complete in order: cnt<=4 => the 4 oldest (current chunk) landed
      asm volatile("s_wait_asynccnt 0x4" ::: "memory");
    } else {
      asm volatile("s_wait_asynccnt 0x0" ::: "memory");
    }
#pragma unroll
    for (int j = 0; j < 4; ++j) {
      const float4 v = buf[cur + j][tid];  // each lane reads only its own 16B
      mn = fminf(mn, fminf(fminf(v.x, v.y), fminf(v.z, v.w)));
      mx = fmaxf(mx, fmaxf(fmaxf(v.x, v.y), fmaxf(v.z, v.w)));
    }
  }

  block_minmax_reduce(mn, mx);
  if (tid == 0) { ws[b] = mn; ws[R_BLOCKS + b] = mx; }
}

// ---------------- stage 2: reduce the 1024 per-block results ----------------
__global__ void __launch_bounds__(256)
minmax_stage2(float* __restrict__ ws) {
  float mn = FLT_MAX, mx = -FLT_MAX;
  for (int i = threadIdx.x; i < R_BLOCKS; i += 256) {
    mn = fminf(mn, ws[i]);
    mx = fmaxf(mx, ws[R_BLOCKS + i]);
  }
  block_minmax_reduce(mn, mx);
  if (threadIdx.x == 0) {
    ws[2 * R_BLOCKS]     = mn;
    ws[2 * R_BLOCKS + 1] = mx;
  }
}

// ---------------- stage 3: per-channel recurrence along n ----------------
// One thread per (l, m) channel: 32768 threads, 64/block -> 512 blocks spread
// across WGPs (only ~1 wave/SIMD exists, so the load pipeline must be deep).
#define C_THREADS 64

__global__ void __launch_bounds__(C_THREADS)
recurrence_kernel(const float* __restrict__ in, const float* __restrict__ alpha,
                  const float* __restrict__ beta, const float* __restrict__ ws,
                  float* __restrict__ out) {
  const int c  = blockIdx.x * C_THREADS + threadIdx.x;   // channel id
  const int l  = c >> 10;                                 // / M_DIM
  const int mi = c & (M_DIM - 1);
  const size_t base = (size_t)l * N_DIM * M_DIM + (size_t)mi;
  const float* px = in  + base;   // x[l, k, mi], stride M_DIM over k
  float*       po = out + base;   // w[l, k, mi], same layout

  const float mn  = ws[2 * R_BLOCKS];
  const float mx  = ws[2 * R_BLOCKS + 1];
  const float inv = 1.0f / (mx - mn);
  const float a   = alpha[0];
  const float b   = beta[0];

  float w = 1.0f;
  __builtin_nontemporal_store(w, po);          // k = 0: w0 = 1
  const int STEPS = N_DIM - 1;                 // 2047 recurrence steps

  int k = 0;
  for (; k + 16 <= STEPS; k += 16) {
    float xs[16];
#pragma unroll
    for (int j = 0; j < 16; ++j) {
      xs[j] = px[(size_t)(k + j) * M_DIM];
      if (k + j + 32 < N_DIM)                  // prefetch 32 planes (128KB) ahead
        __builtin_prefetch(px + (size_t)(k + j + 32) * M_DIM, 0, 1);
    }
#pragma unroll
    for (int j = 0; j < 16; ++j) {
      const float xn = (xs[j] - mn) * inv;
      float r = __builtin_amdgcn_rcpf(w);      // v_rcp_f32
      r = r * fmaf(-w, r, 2.0f);               // 1 Newton step: ~0.5 ulp of 1/w
      w = fmaf(a, r, fmaf(-b, xn, w));         // w += a/w - b*xn
      __builtin_nontemporal_store(w, po + (size_t)(k + j + 1) * M_DIM);
    }
  }
  for (; k < STEPS; ++k) {                     // tail (15 steps)
    const float x  = px[(size_t)k * M_DIM];
    const float xn = (x - mn) * inv;
    float r = __builtin_amdgcn_rcpf(w);
    r = r * fmaf(-w, r, 2.0f);
    w = fmaf(a, r, fmaf(-b, xn, w));
    __builtin_nontemporal_store(w, po + (size_t)(k + 1) * M_DIM);
  }
}

// ---------------- host entry ----------------
extern "C" void kernel_launch(void* const* d_in, const int* in_sizes, int n_in,
                              void* d_out, int out_size, void* d_ws, size_t ws_size,
                              hipStream_t stream) {
  (void)in_sizes; (void)n_in; (void)out_size; (void)ws_size;
  const float* in    = (const float*)d_in[0];
  const float* alpha = (const float*)d_in[1];
  const float* beta  = (const float*)d_in[2];
  float* out = (float*)d_out;
  float* ws  = (float*)d_ws;   // [0..1023] blk mins, [1024..2047] blk maxs, [2048..2049] final

  minmax_stage1<<<R_BLOCKS, R_THREADS, 0, stream>>>(in, ws);
  minmax_stage2<<<1, 256, 0, stream>>>(ws);
  recurrence_kernel<<<(L_DIM * M_DIM) / C_THREADS, C_THREADS, 0, stream>>>(
      in, alpha, beta, ws, out);
}